// BioGCN_81552839016828
// MI455X (gfx1250) — compile-verified
//
#include <hip/hip_runtime.h>
#include <stdint.h>

#define N_    16
#define M_    20000
#define FIN_  64
#define K_    4
#define NNZ_  320000
#define KM_   (K_ * M_)        // 80000
#define ROW_  1024             // FIN_ * N_
#define CHUNK_ 8

typedef unsigned int u32;
typedef u32   v4u __attribute__((ext_vector_type(4)));
typedef int   v8i __attribute__((ext_vector_type(8)));
typedef int   v4i __attribute__((ext_vector_type(4)));
typedef float v4f __attribute__((ext_vector_type(4)));

#if __has_builtin(__builtin_amdgcn_tensor_load_to_lds)
#define USE_TDM 1
#endif

// ---------------- Kernel 1: x0g[m, f*16+n] = x[n, m, f] ----------------
// x is read exactly once -> NT loads (don't evict the L2-resident x0g).
// x0g is the 82MB gather target reused 64x from L2 -> regular (RT) stores.
__global__ __launch_bounds__(256) void build_x0(const float* __restrict__ x,
                                                float* __restrict__ x0g) {
  __shared__ float tile[16 * 65];           // padded: bank-conflict free both ways
  const int m = blockIdx.x, t = threadIdx.x;
#pragma unroll
  for (int j = 0; j < 4; ++j) {
    const int idx = t + 256 * j;
    const int n = idx >> 6, f = idx & 63;   // coalesced 256B reads per 64 lanes
    tile[n * 65 + f] =
        __builtin_nontemporal_load(x + ((size_t)n * M_ + m) * FIN_ + f);
  }
  __syncthreads();
#pragma unroll
  for (int j = 0; j < 4; ++j) {
    const int p = t + 256 * j;              // p = f*16 + n
    const int f = p >> 4, n = p & 15;
    x0g[(size_t)m * ROW_ + p] = tile[n * 65 + f];   // coalesced 4KB row write (RT)
  }
}

// ---------------- CSR build ----------------
__global__ __launch_bounds__(256) void zero_counts(int* __restrict__ cnt) {
  const int i = blockIdx.x * 256 + threadIdx.x;
  if (i < KM_) cnt[i] = 0;
}

__global__ __launch_bounds__(256) void count_rows(const int* __restrict__ rows,
                                                  int* __restrict__ cnt) {
  const int i = blockIdx.x * 256 + threadIdx.x;
  if (i < K_ * NNZ_) {
    const int k = i / NNZ_;
    atomicAdd(&cnt[k * M_ + rows[i]], 1);
  }
}

__global__ __launch_bounds__(256) void scan_blocks(const int* __restrict__ cnt,
                                                   int* __restrict__ ptr,
                                                   int* __restrict__ bsums) {
  __shared__ int s[256];
  const int t = threadIdx.x;
  const int base = blockIdx.x * 1024 + t * 4;
  int v[4], pre[4];
#pragma unroll
  for (int j = 0; j < 4; ++j) v[j] = (base + j < KM_) ? cnt[base + j] : 0;
  pre[0] = 0; pre[1] = v[0]; pre[2] = pre[1] + v[1]; pre[3] = pre[2] + v[2];
  const int tsum = pre[3] + v[3];
  s[t] = tsum;
  __syncthreads();
  for (int off = 1; off < 256; off <<= 1) {
    const int add = (t >= off) ? s[t - off] : 0;
    __syncthreads();
    s[t] += add;
    __syncthreads();
  }
  const int excl = s[t] - tsum;
#pragma unroll
  for (int j = 0; j < 4; ++j)
    if (base + j < KM_) ptr[base + j] = excl + pre[j];
  if (t == 255) bsums[blockIdx.x] = s[255];
}

__global__ __launch_bounds__(256) void scan_bsums(int* __restrict__ bsums, int nb) {
  __shared__ int s[256];
  const int t = threadIdx.x;
  const int v = (t < nb) ? bsums[t] : 0;
  s[t] = v;
  __syncthreads();
  for (int off = 1; off < 256; off <<= 1) {
    const int add = (t >= off) ? s[t - off] : 0;
    __syncthreads();
    s[t] += add;
    __syncthreads();
  }
  if (t < nb) bsums[t] = s[t] - v;          // exclusive
}

__global__ __launch_bounds__(256) void add_offsets(int* __restrict__ ptr,
                                                   const int* __restrict__ bsums,
                                                   int* __restrict__ cnt) {
  const int i = blockIdx.x * 256 + threadIdx.x;
  if (i < KM_) { ptr[i] += bsums[i >> 10]; cnt[i] = 0; }  // cnt reused as fill cursor
  if (i == 0) ptr[KM_] = K_ * NNZ_;
}

__global__ __launch_bounds__(256) void fill_csr(const float* __restrict__ vals,
                                                const int* __restrict__ rows,
                                                const int* __restrict__ cols,
                                                const int* __restrict__ ptr,
                                                int* __restrict__ cur,
                                                int* __restrict__ ccols,
                                                float* __restrict__ cvals) {
  const int i = blockIdx.x * 256 + threadIdx.x;
  if (i < K_ * NNZ_) {
    const int k = i / NNZ_;
    const int r = k * M_ + rows[i];
    const int pos = atomicAdd(&cur[r], 1);
    const int idx = ptr[r] + pos;
    ccols[idx] = cols[i];
    cvals[idx] = vals[i];
  }
}

// ---------------- TDM gather-mode descriptor issue (wave 0 only) ----------------
#if defined(USE_TDM)
__device__ __forceinline__ void tdm_gather(u32 ldsOff, const float* __restrict__ x0g,
                                           const int* __restrict__ cols, int cnt) {
  const unsigned long long ga = (unsigned long long)(uintptr_t)x0g;
  v4u g0;
  g0[0] = 1u | (1u << 31);                      // count=1, gather_mode=1, 16-bit idx
  g0[1] = ldsOff;                               // lds_addr (bytes)
  g0[2] = (u32)(ga & 0xFFFFFFFFull);            // global_addr[31:0]
  g0[3] = (u32)((ga >> 32) & 0x1FFFFFFull) | (2u << 30);  // addr[56:32] | type=2
  v8i g1;
  g1[0] = (int)(2u << 16);                      // data_size = 4 bytes
  g1[1] = (int)(((u32)ROW_ & 0xFFFFu) << 16);   // tensor_dim0[15:0]
  g1[2] = (int)(((u32)ROW_ >> 16) | (((u32)M_ & 0xFFFFu) << 16)); // dim0 hi | dim1 lo
  g1[3] = (int)(((u32)M_ >> 16) | ((u32)ROW_ << 16));             // dim1 hi | tile_dim0
  g1[4] = cnt;                                  // tile_dim1 = #valid indices
  g1[5] = ROW_;                                 // tensor_dim0_stride[31:0]
  g1[6] = 0;
  g1[7] = 0;
  int c[8];
#pragma unroll
  for (int i = 0; i < 8; ++i) c[i] = (i < cnt) ? cols[i] : 0;   // uniform scalar loads
  v4i g2 = { c[0] | (c[1] << 16), c[2] | (c[3] << 16),
             c[4] | (c[5] << 16), c[6] | (c[7] << 16) };
  v4i g3 = { 0, 0, 0, 0 };
  v8i g4 = { 0, 0, 0, 0, 0, 0, 0, 0 };          // extra operand of 6-arg builtin
  __builtin_amdgcn_tensor_load_to_lds(g0, g1, g2, g3, g4, 0);
}
#endif

// ---------------- Kernel 7: SpMM + bias + relu + transposed store ----------------
__global__ __launch_bounds__(256) void spmm_kernel(const float* __restrict__ x0g,
                                                   const int* __restrict__ rowptr,
                                                   const int* __restrict__ ccols,
                                                   const float* __restrict__ cvals,
                                                   float* __restrict__ out) {
  __shared__ __align__(16) float stage[CHUNK_ * ROW_];   // 32KB, reused for output staging
  const int m = blockIdx.x, t = threadIdx.x;
  v4f acc[K_];
#pragma unroll
  for (int k = 0; k < K_; ++k) acc[k] = (v4f){0.f, 0.f, 0.f, 0.f};

  const v4f x0r = ((const v4f*)(x0g + (size_t)m * ROW_))[t];
#if defined(USE_TDM)
  const u32 ldsOff = (u32)(uintptr_t)(&stage[0]);
#endif

  for (int k = 0; k < K_; ++k) {
    const int beg = rowptr[k * M_ + m];
    const int end = rowptr[k * M_ + m + 1];
    for (int c = beg; c < end; c += CHUNK_) {
      const int cnt = min(CHUNK_, end - c);
#if defined(USE_TDM)
      if (t < 32) tdm_gather(ldsOff, x0g, ccols + c, cnt);  // wave 0 issues DMA gather
#else
      for (int r = 0; r < CHUNK_; ++r) {          // fallback: plain staged gather
        if (r < cnt) {
          const int col = ccols[c + r];
          ((v4f*)stage)[r * 256 + t] = ((const v4f*)(x0g + (size_t)col * ROW_))[t];
        }
      }
#endif
      float v[CHUNK_];                            // uniform scalar loads overlap the DMA
#pragma unroll
      for (int i = 0; i < CHUNK_; ++i) v[i] = (i < cnt) ? cvals[c + i] : 0.f;
#if defined(USE_TDM)
      if (t < 32) __builtin_amdgcn_s_wait_tensorcnt(0);
#endif
      __syncthreads();                            // gathered rows visible to all waves
      const v4f* sg = (const v4f*)stage;
#pragma unroll
      for (int r = 0; r < CHUNK_; ++r) {
        if (r < cnt) acc[k] += v[r] * sg[r * 256 + t];   // fused to v_fma/v_pk_fma
      }
      __syncthreads();                            // done reading before next DMA reuses stage
    }
  }

  // epilogue: y = relu(acc + x0_row); stage in LDS, then coalesced NT b128 stores
#pragma unroll
  for (int k = 0; k < K_; ++k) {
    v4f y = acc[k] + x0r;
    y.x = fmaxf(y.x, 0.f); y.y = fmaxf(y.y, 0.f);
    y.z = fmaxf(y.z, 0.f); y.w = fmaxf(y.w, 0.f);
    ((v4f*)stage)[k * 256 + t] = y;               // stage[k][p], p = 4t..4t+3
  }
  __syncthreads();
#pragma unroll
  for (int j = 0; j < 4; ++j) {
    const int n = (t >> 6) + 4 * j;               // 0..15
    const int f = t & 63;                         // 0..63
    const int p = f * 16 + n;
    const v4f o = { stage[0 * ROW_ + p], stage[1 * ROW_ + p],
                    stage[2 * ROW_ + p], stage[3 * ROW_ + p] };
    // 328MB write-once stream: NT so it doesn't evict the L2-resident x0g
    __builtin_nontemporal_store(
        o, (v4f*)(out + ((size_t)n * M_ + m) * (FIN_ * K_)) + f);  // out[n,m,f*4+k]
  }
}

// ---------------- launch ----------------
extern "C" void kernel_launch(void* const* d_in, const int* in_sizes, int n_in,
                              void* d_out, int out_size, void* d_ws, size_t ws_size,
                              hipStream_t stream) {
  (void)in_sizes; (void)n_in; (void)out_size; (void)ws_size;
  const float* x      = (const float*)d_in[0];
  const float* L_vals = (const float*)d_in[1];
  const int*   L_rows = (const int*)d_in[2];
  const int*   L_cols = (const int*)d_in[3];
  float* out = (float*)d_out;

  char* w = (char*)d_ws;                       // ws layout (~92.8 MB total)
  float* x0g   = (float*)(w);                  // 81,920,000 B : X0 [M,1024]
  int*   cnt   = (int*)  (w + 81920000);       //    320,000 B : row counts / cursors
  int*   ptr   = (int*)  (w + 82240000);       //    320,016 B : CSR row ptr (KM_+1)
  int*   ccols = (int*)  (w + 82560016);       //  5,120,000 B : CSR cols
  float* cvals = (float*)(w + 87680016);       //  5,120,000 B : CSR vals
  int*   bsums = (int*)  (w + 92800016);       //      1,024 B : scan partials

  build_x0<<<M_, 256, 0, stream>>>(x, x0g);
  zero_counts<<<(KM_ + 255) / 256, 256, 0, stream>>>(cnt);
  count_rows<<<(K_ * NNZ_ + 255) / 256, 256, 0, stream>>>(L_rows, cnt);
  const int nb = (KM_ + 1023) / 1024;          // 79 scan blocks
  scan_blocks<<<nb, 256, 0, stream>>>(cnt, ptr, bsums);
  scan_bsums<<<1, 256, 0, stream>>>(bsums, nb);
  add_offsets<<<(KM_ + 255) / 256, 256, 0, stream>>>(ptr, bsums, cnt);
  fill_csr<<<(K_ * NNZ_ + 255) / 256, 256, 0, stream>>>(L_vals, L_rows, L_cols,
                                                        ptr, cnt, ccols, cvals);
  spmm_kernel<<<M_, 256, 0, stream>>>(x0g, ptr, ccols, cvals, out);
}